// WeightedMSELoss_8753143349671
// MI455X (gfx1250) — compile-verified
//
#include <hip/hip_runtime.h>
#include <hip/hip_bf16.h>

// ---------------------------------------------------------------------------
// WeightedMSELoss (DeepVO-style) for MI455X / gfx1250.
//
// Bandwidth-bound: ~201 MB streamed once -> use b128 non-temporal loads.
// Deterministic two-pass reduction; final 32-lane reduction done on the
// matrix unit via v_wmma_f32_16x16x4_f32 (B = ones => D rows = row sums).
// ---------------------------------------------------------------------------

typedef __attribute__((ext_vector_type(4))) float v4f;
typedef __attribute__((ext_vector_type(2))) float v2f;
typedef __attribute__((ext_vector_type(8))) float v8f;

#define MAIN_BLOCKS  2048
#define MAIN_THREADS 256

__device__ __forceinline__ float wrap_angle(float a) {
    // Matches jnp.where(a > pi, a - 2pi, a); jnp.where(a < -pi, a + 2pi, a)
    const float PI_F     = 3.14159274101257324f;   // (float)pi
    const float TWO_PI_F = 6.28318548202514648f;   // (float)(2*pi)
    a = (a >  PI_F) ? a - TWO_PI_F : a;
    a = (a < -PI_F) ? a + TWO_PI_F : a;
    return a;
}

// Pass 1: each thread grid-strides over groups of 12 floats (= 2 complete
// 6-channel poses). 3x b128 NT loads per tensor per group. Per-block partial
// sums of squared diffs (trans, rot) written to workspace.
__global__ __launch_bounds__(MAIN_THREADS)
void vo_loss_partial(const float* __restrict__ pred,
                     const float* __restrict__ target,
                     float* __restrict__ partials,
                     int nGroups) {
    float ts = 0.0f, rs = 0.0f;
    const int tid    = threadIdx.x;
    const int stride = gridDim.x * blockDim.x;

    for (int g = blockIdx.x * blockDim.x + tid; g < nGroups; g += stride) {
        const v4f* p = (const v4f*)(pred   + (size_t)g * 12);
        const v4f* t = (const v4f*)(target + (size_t)g * 12);
        // Streaming data (201 MB > 192 MB L2, touched once): non-temporal.
        v4f p0 = __builtin_nontemporal_load(p);
        v4f p1 = __builtin_nontemporal_load(p + 1);
        v4f p2 = __builtin_nontemporal_load(p + 2);
        v4f t0 = __builtin_nontemporal_load(t);
        v4f t1 = __builtin_nontemporal_load(t + 1);
        v4f t2 = __builtin_nontemporal_load(t + 2);

        float d;
        // Translation channels: flat indices 0,1,2 and 6,7,8
        d = p0.x - t0.x; ts = fmaf(d, d, ts);
        d = p0.y - t0.y; ts = fmaf(d, d, ts);
        d = p0.z - t0.z; ts = fmaf(d, d, ts);
        d = p1.z - t1.z; ts = fmaf(d, d, ts);
        d = p1.w - t1.w; ts = fmaf(d, d, ts);
        d = p2.x - t2.x; ts = fmaf(d, d, ts);
        // Rotation channels (angle-wrapped): flat indices 3,4,5 and 9,10,11
        d = wrap_angle(p0.w) - wrap_angle(t0.w); rs = fmaf(d, d, rs);
        d = wrap_angle(p1.x) - wrap_angle(t1.x); rs = fmaf(d, d, rs);
        d = wrap_angle(p1.y) - wrap_angle(t1.y); rs = fmaf(d, d, rs);
        d = wrap_angle(p2.y) - wrap_angle(t2.y); rs = fmaf(d, d, rs);
        d = wrap_angle(p2.z) - wrap_angle(t2.z); rs = fmaf(d, d, rs);
        d = wrap_angle(p2.w) - wrap_angle(t2.w); rs = fmaf(d, d, rs);
    }

    __shared__ float sT[MAIN_THREADS];
    __shared__ float sR[MAIN_THREADS];
    sT[tid] = ts;
    sR[tid] = rs;
    __syncthreads();
    for (int off = MAIN_THREADS / 2; off > 0; off >>= 1) {
        if (tid < off) {
            sT[tid] += sT[tid + off];
            sR[tid] += sR[tid + off];
        }
        __syncthreads();
    }
    if (tid == 0) {
        partials[2 * blockIdx.x + 0] = sT[0];
        partials[2 * blockIdx.x + 1] = sR[0];
    }
}

// Sum one float per lane across the full wave32 using the matrix unit.
// A (16x4): lane L (<16) holds A[L][0]=v_L, A[L][1]=0; lane L+16 holds
// A[L][2]=v_{L+16}, A[L][3]=0. B = ones(4x16) => D[M][N] = v_M + v_{M+16}.
// C/D layout: VGPR r, lanes 0-15 hold M=r, lanes 16-31 hold M=8+r. So
// sum(d[0..7]) on lane0 covers M=0..7, on lane16 covers M=8..15.
__device__ __forceinline__ float wave_reduce_wmma(float v) {
    v2f a;  a.x = v;    a.y = 0.0f;
    v2f b;  b.x = 1.0f; b.y = 1.0f;
    v8f c = {};
    c = __builtin_amdgcn_wmma_f32_16x16x4_f32(
        /*neg_a=*/false, a, /*neg_b=*/false, b,
        /*c_mod=*/(short)0, c, /*reuse_a=*/false, /*reuse_b=*/false);
    float s = c[0] + c[1] + c[2] + c[3] + c[4] + c[5] + c[6] + c[7];
    return __shfl(s, 0, 32) + __shfl(s, 16, 32);
}

// Pass 2: one wave. Lanes stride the per-block partials (uniform trip count,
// EXEC all-ones before the WMMAs), then WMMA-reduce and write the 3 outputs.
__global__ __launch_bounds__(32)
void vo_loss_final(const float* __restrict__ partials,
                   float* __restrict__ out,
                   int numPartials,
                   float invCount) {
    const int lane = threadIdx.x;
    float ts = 0.0f, rs = 0.0f;
    for (int i = lane; i < numPartials; i += 32) {
        ts += partials[2 * i + 0];
        rs += partials[2 * i + 1];
    }
    float transSum = wave_reduce_wmma(ts);
    float rotSum   = wave_reduce_wmma(rs);
    if (lane == 0) {
        float transLoss = transSum * invCount * 1.0f;    // TRANS_WEIGHT
        float rotLoss   = rotSum   * invCount * 100.0f;  // ROT_WEIGHT
        out[0] = transLoss + rotLoss;  // total_loss
        out[1] = transLoss;            // trans_loss
        out[2] = rotLoss;              // rot_loss
    }
}

extern "C" void kernel_launch(void* const* d_in, const int* in_sizes, int n_in,
                              void* d_out, int out_size, void* d_ws, size_t ws_size,
                              hipStream_t stream) {
    const float* pred   = (const float*)d_in[0];
    const float* target = (const float*)d_in[1];
    float* out = (float*)d_out;
    float* ws  = (float*)d_ws;   // 2 floats per block = 16 KB

    const long long N = (long long)in_sizes[0];      // B*T*6 = 25,165,824
    const int nGroups = (int)(N / 12);               // 12-float groups (2 poses)
    const float invCount = 1.0f / (float)(N / 2);    // elements per category (3 of 6 ch)

    vo_loss_partial<<<MAIN_BLOCKS, MAIN_THREADS, 0, stream>>>(pred, target, ws, nGroups);
    vo_loss_final<<<1, 32, 0, stream>>>(ws, out, MAIN_BLOCKS, invCount);
}